// RiemannianPooling_32323923870234
// MI455X (gfx1250) — compile-verified
//
#include <hip/hip_runtime.h>

typedef float v2f __attribute__((ext_vector_type(2)));
typedef float v8f __attribute__((ext_vector_type(8)));

#define NN 32
#define PP 3200            // M*T*V
#define CC 16
#define DD 136             // C(C+1)/2
#define DPAD 144           // padded D (9 tiles of 16)
#define LSTR 17            // LDS row stride for 16x16 matrices (bank-conflict pad)
#define EPSF 1e-4f
#define SQRT2F 1.41421356237309514547f

// Wave-local LDS fence: all LDS data in the Jacobi solver is private to one
// wave32, whose lanes run in lockstep. Once this wave's DS ops have retired
// (DScnt==0) every lane sees them -- no workgroup barrier needed. The
// wave_barrier()s pin compiler scheduling; the asm has a memory clobber.
__device__ __forceinline__ void wave_sync_lds() {
  __builtin_amdgcn_wave_barrier();
  asm volatile("s_wait_dscnt 0x0" ::: "memory");
  __builtin_amdgcn_wave_barrier();
}

// ---------------------------------------------------------------------------
// Wave-cooperative parallel cyclic Jacobi eigensolver for a symmetric 16x16
// matrix held in LDS (row stride LSTR). One wave32 per matrix. On exit:
// Am diagonal holds eigenvalues (junk elsewhere), Vm holds eigenvectors
// (columns), cs[0..15] holds log(clamp(eigval, EPS)).
// Wave-local sync only: independent waves in a block proceed unsynchronized.
// ---------------------------------------------------------------------------
__device__ void jacobi_logm_16(float* Am, float* Vm, float* cs, int lane) {
  // V = I
  for (int it = 0; it < 8; ++it) {
    int e = lane + 32 * it, r = e >> 4, c = e & 15;
    Vm[r * LSTR + c] = (r == c) ? 1.0f : 0.0f;
  }
  wave_sync_lds();
  for (int sw = 0; sw < 9; ++sw) {
    for (int rd = 0; rd < 15; ++rd) {
      // 8 disjoint pairs (round-robin tournament over 16 indices)
      if (lane < 8) {
        int p, q;
        if (lane == 0) { p = 15; q = rd % 15; }
        else { p = (lane + rd) % 15; q = (15 - lane + rd) % 15; }
        float app = Am[p * LSTR + p], aqq = Am[q * LSTR + q], apq = Am[p * LSTR + q];
        float cc = 1.0f, ss = 0.0f;
        if (fabsf(apq) > 1e-12f) {
          float tau = (aqq - app) / (2.0f * apq);
          float tt = ((tau >= 0.0f) ? 1.0f : -1.0f) / (fabsf(tau) + sqrtf(1.0f + tau * tau));
          cc = 1.0f / sqrtf(1.0f + tt * tt);
          ss = tt * cc;
        }
        cs[lane * 2] = cc; cs[lane * 2 + 1] = ss;
      }
      wave_sync_lds();
      // row rotations: A <- J^T A  (each lane owns a (pair, col) slot)
      for (int it = 0; it < 4; ++it) {
        int e = lane + 32 * it;
        int k = e >> 4, col = e & 15;
        int p, q;
        if (k == 0) { p = 15; q = rd % 15; }
        else { p = (k + rd) % 15; q = (15 - k + rd) % 15; }
        float cc = cs[k * 2], ss = cs[k * 2 + 1];
        float ap = Am[p * LSTR + col], aq = Am[q * LSTR + col];
        Am[p * LSTR + col] = cc * ap - ss * aq;
        Am[q * LSTR + col] = ss * ap + cc * aq;
      }
      wave_sync_lds();
      // col rotations: A <- A J, V <- V J
      for (int it = 0; it < 4; ++it) {
        int e = lane + 32 * it;
        int k = e >> 4, row = e & 15;
        int p, q;
        if (k == 0) { p = 15; q = rd % 15; }
        else { p = (k + rd) % 15; q = (15 - k + rd) % 15; }
        float cc = cs[k * 2], ss = cs[k * 2 + 1];
        float ap = Am[row * LSTR + p], aq = Am[row * LSTR + q];
        Am[row * LSTR + p] = cc * ap - ss * aq;
        Am[row * LSTR + q] = ss * ap + cc * aq;
        float vp = Vm[row * LSTR + p], vq = Vm[row * LSTR + q];
        Vm[row * LSTR + p] = cc * vp - ss * vq;
        Vm[row * LSTR + q] = ss * vp + cc * vq;
      }
      wave_sync_lds();
    }
  }
  if (lane < 16) {
    float w = Am[lane * LSTR + lane];
    cs[lane] = logf(fmaxf(w, EPSF));
  }
  wave_sync_lds();
}

// ---------------------------------------------------------------------------
// Prep: eigenvalues of W_sym (serial tiny Jacobi), sorted |lambda| = singular
// values; build diff_exp factor F[16][16]; zero the mean accumulator.
// sF layout: [0..15] = s, [16..271] = F row-major.
// ---------------------------------------------------------------------------
__global__ __launch_bounds__(256) void prep_kernel(const float* __restrict__ W,
                                                   float* __restrict__ sF,
                                                   float* __restrict__ meansum) {
  __shared__ float A[16][16];
  __shared__ float sv[16];
  int t = threadIdx.x;
  for (int i = t; i < NN * 256; i += 256) meansum[i] = 0.0f;
  {
    int r = t >> 4, c = t & 15;
    A[r][c] = 0.5f * (W[r * 16 + c] + W[c * 16 + r]);
  }
  __syncthreads();
  if (t == 0) {
    for (int sw = 0; sw < 12; ++sw)
      for (int p = 0; p < 16; ++p)
        for (int q = p + 1; q < 16; ++q) {
          float apq = A[p][q];
          if (fabsf(apq) < 1e-12f) continue;
          float app = A[p][p], aqq = A[q][q];
          float tau = (aqq - app) / (2.0f * apq);
          float tt = ((tau >= 0.0f) ? 1.0f : -1.0f) / (fabsf(tau) + sqrtf(1.0f + tau * tau));
          float cc = 1.0f / sqrtf(1.0f + tt * tt), ss = tt * cc;
          for (int i = 0; i < 16; ++i) {
            float aip = A[i][p], aiq = A[i][q];
            A[i][p] = cc * aip - ss * aiq;
            A[i][q] = ss * aip + cc * aiq;
          }
          for (int i = 0; i < 16; ++i) {
            float api = A[p][i], aqi = A[q][i];
            A[p][i] = cc * api - ss * aqi;
            A[q][i] = ss * api + cc * aqi;
          }
        }
    for (int i = 0; i < 16; ++i) sv[i] = fabsf(A[i][i]);
    for (int i = 0; i < 16; ++i)        // sort descending (SVD order)
      for (int j = i + 1; j < 16; ++j)
        if (sv[j] > sv[i]) { float tmp = sv[i]; sv[i] = sv[j]; sv[j] = tmp; }
    for (int i = 0; i < 16; ++i) sF[i] = sv[i];
  }
  __syncthreads();
  {
    int i = t >> 4, j = t & 15;
    float si = sv[i], sj = sv[j];
    float den = si - sj;
    bool eq = fabsf(den) < EPSF;
    float esi = expf(si), esj = expf(sj);
    float num = eq ? 0.5f * (esi + esj) : (esi - esj);
    sF[16 + t] = num / (eq ? 1.0f : den);
  }
}

// ---------------------------------------------------------------------------
// Bulk sym_logm: 102,400 16x16 eigendecompositions. One matrix per wave,
// 8 per block, wave-local sync in the solver. Writes logx and accumulates
// per-n mean sums via f32 atomics after one block-level LDS reduction
// (the single cross-wave barrier in this kernel).
// ---------------------------------------------------------------------------
__global__ __launch_bounds__(256) void eig_logm_kernel(const float* __restrict__ x,
                                                       float* __restrict__ logx,
                                                       float* __restrict__ meansum) {
  __shared__ float sA[8][16 * LSTR];
  __shared__ float sV[8][16 * LSTR];
  __shared__ float scs[8][16];
  int t = threadIdx.x;
  int w = t >> 5, lane = t & 31;
  long blk = blockIdx.x;                 // 12800 blocks
  int n = (int)(blk / 400);
  int p0 = (int)(blk % 400) * 8;
  long mat = (long)n * PP + p0 + w;
  const float* xin = x + mat * 256;
  float* Am = sA[w];
  float* Vm = sV[w];
  float* cs = scs[w];
  for (int it = 0; it < 8; ++it) {
    int e = lane + 32 * it;
    Am[(e >> 4) * LSTR + (e & 15)] = xin[e];
  }
  wave_sync_lds();
  jacobi_logm_16(Am, Vm, cs, lane);
  // logm = V diag(log w) V^T
  float outv[8];
  for (int it = 0; it < 8; ++it) {
    int e = lane + 32 * it, r = e >> 4, c = e & 15;
    float acc = 0.0f;
    for (int k = 0; k < 16; ++k) acc += Vm[r * LSTR + k] * cs[k] * Vm[c * LSTR + k];
    outv[it] = acc;
  }
  wave_sync_lds();
  for (int it = 0; it < 8; ++it) {
    int e = lane + 32 * it;
    Am[(e >> 4) * LSTR + (e & 15)] = outv[it];
    logx[mat * 256 + e] = outv[it];
  }
  __syncthreads();   // cross-wave: all 8 logm matrices visible block-wide
  // block-level partial sum over 8 matrices (all same n), then one atomic each
  {
    int r = t >> 4, c = t & 15;
    float s = 0.0f;
    for (int ww = 0; ww < 8; ++ww) s += sA[ww][r * LSTR + c];
    atomicAdd(&meansum[n * 256 + t], s);
  }
}

// ---------------------------------------------------------------------------
// Per-n logm of the mean matrix, then lower_triangle (sqrt2 off-diag).
// One wave per block, 32 blocks. Writes xm[n][DPAD] (zero-padded).
// ---------------------------------------------------------------------------
__global__ __launch_bounds__(32) void mean_logm_kernel(const float* __restrict__ meansum,
                                                       float* __restrict__ xm) {
  __shared__ float Am[16 * LSTR];
  __shared__ float Vm[16 * LSTR];
  __shared__ float cs[16];
  int lane = threadIdx.x;
  int n = blockIdx.x;
  const float invP = 1.0f / (float)PP;
  for (int it = 0; it < 8; ++it) {
    int e = lane + 32 * it;
    Am[(e >> 4) * LSTR + (e & 15)] = meansum[n * 256 + e] * invP;
  }
  wave_sync_lds();
  jacobi_logm_16(Am, Vm, cs, lane);
  for (int d = lane; d < DPAD; d += 32) {
    float val = 0.0f;
    if (d < DD) {
      int r = 0;
      while ((r + 1) * (r + 2) / 2 <= d) r++;
      int c = d - r * (r + 1) / 2;
      float acc = 0.0f;
      for (int k = 0; k < 16; ++k) acc += Vm[r * LSTR + k] * cs[k] * Vm[c * LSTR + k];
      val = acc * ((r == c) ? 1.0f : SQRT2F);
    }
    xm[n * DPAD + d] = val;
  }
}

// ---------------------------------------------------------------------------
// Center + diff_exp factor + lower-triangle vectorize: xc[n][p][DPAD].
// 2 matrices per 288-thread block (144 threads each -> one d per thread).
// ---------------------------------------------------------------------------
__global__ __launch_bounds__(288) void center_kernel(const float* __restrict__ logx,
                                                     const float* __restrict__ meansum,
                                                     const float* __restrict__ sF,
                                                     float* __restrict__ xc) {
  int t = threadIdx.x;
  int sub = t / DPAD;
  int d = t % DPAD;
  long mat = (long)blockIdx.x * 2 + sub;     // 51200 blocks
  int n = (int)(mat / PP);
  float val = 0.0f;
  if (d < DD) {
    int r = 0;
    while ((r + 1) * (r + 2) / 2 <= d) r++;
    int c = d - r * (r + 1) / 2;
    int rc = r * 16 + c;
    float mean = meansum[n * 256 + rc] * (1.0f / (float)PP);
    float F = sF[16 + rc];
    float sc = (r == c) ? 1.0f : SQRT2F;
    val = (logx[mat * 256 + rc] - mean) * F * sc;
  }
  xc[mat * DPAD + d] = val;
}

// ---------------------------------------------------------------------------
// Gram matrix via fp32 WMMA: xcov[n] = xc[n]^T @ xc[n] over K=3200.
// Block = 9 waves; wave j owns output tile (blockIdx.x, j) of the 9x9 tile
// grid. K staged through LDS in 32-row chunks (float4 global loads).
// ---------------------------------------------------------------------------
__global__ __launch_bounds__(288) void gram_kernel(const float* __restrict__ xc,
                                                   float* __restrict__ xcov) {
  __shared__ float tile[32][148];
  int t = threadIdx.x, w = t >> 5, lane = t & 31;
  int i0 = blockIdx.x * 16;
  int n = blockIdx.y;
  const float* base = xc + (long)n * PP * DPAD;
  int j0 = w * 16;
  int m = lane & 15;              // M/N index of the fragment row/col
  int kb = (lane >> 4) * 2;       // K split across lane halves
  v8f acc = {0.f, 0.f, 0.f, 0.f, 0.f, 0.f, 0.f, 0.f};
  for (int pc = 0; pc < PP; pc += 32) {
    __syncthreads();
    for (int it = 0; it < 4; ++it) {
      int idx = t + 288 * it;                 // 0..1151 float4 slots
      int row = idx / 36, col4 = idx % 36;
      const float4* src = (const float4*)(base + (long)(pc + row) * DPAD) + col4;
      float4 v = *src;
      __builtin_prefetch(src + 1152, 0, 1);   // next K-chunk -> global_prefetch
      float* dst = &tile[row][col4 * 4];
      dst[0] = v.x; dst[1] = v.y; dst[2] = v.z; dst[3] = v.w;
    }
    __syncthreads();
    for (int kk = 0; kk < 32; kk += 4) {
      v2f a, b;
      a.x = tile[kk + kb][i0 + m];
      a.y = tile[kk + kb + 1][i0 + m];
      b.x = tile[kk + kb][j0 + m];
      b.y = tile[kk + kb + 1][j0 + m];
      acc = __builtin_amdgcn_wmma_f32_16x16x4_f32(false, a, false, b,
                                                  (short)0, acc, false, false);
    }
  }
  int mb = (lane >> 4) * 8;
  float* dst = xcov + ((long)n * DPAD + i0) * DPAD + j0;
  for (int v = 0; v < 8; ++v) dst[(mb + v) * DPAD + m] = acc[v];
}

// ---------------------------------------------------------------------------
// Per-n: scale/symmetrize covariance, Cholesky (136x136), assemble the
// (137x137) block matrix, trace-normalize, write zero-padded 144x144.
// Dynamic LDS: (137*138 + 2) floats.
// ---------------------------------------------------------------------------
__global__ __launch_bounds__(256) void chol_kernel(const float* __restrict__ xcov,
                                                   const float* __restrict__ xm,
                                                   float* __restrict__ outp) {
  extern __shared__ float Lsh[];
  const int STR = 138;
  int t = threadIdx.x, n = blockIdx.x;
  for (int idx = t; idx < 137 * STR; idx += 256) Lsh[idx] = 0.0f;
  __syncthreads();
  const float* cov = xcov + (long)n * DPAD * DPAD;
  const float inv = 1.0f / (float)(PP - 1);
  for (int idx = t; idx < DD * DD; idx += 256) {
    int i = idx / DD, j = idx % DD;
    if (j <= i) Lsh[i * STR + j] = 0.5f * (cov[i * DPAD + j] + cov[j * DPAD + i]) * inv;
  }
  __syncthreads();
  for (int j = 0; j < DD; ++j) {
    if (t == 0) {
      float s = Lsh[j * STR + j];
      for (int k = 0; k < j; ++k) { float v = Lsh[j * STR + k]; s -= v * v; }
      Lsh[j * STR + j] = sqrtf(fmaxf(s, 1e-20f));
    }
    __syncthreads();
    float dj = Lsh[j * STR + j];
    for (int i = j + 1 + t; i < DD; i += 256) {
      float s = Lsh[i * STR + j];
      for (int k = 0; k < j; ++k) s -= Lsh[i * STR + k] * Lsh[j * STR + k];
      Lsh[i * STR + j] = s / dj;
    }
    __syncthreads();
  }
  for (int d = t; d < DD; d += 256) Lsh[DD * STR + d] = xm[n * DPAD + d];
  if (t == 0) Lsh[DD * STR + DD] = 1.0f;
  __syncthreads();
  if (t == 0) {
    float tr = 0.0f;
    for (int j = 0; j < 137; ++j) tr += Lsh[j * STR + j];
    Lsh[137 * STR] = tr;
  }
  __syncthreads();
  float tr = Lsh[137 * STR];
  float* dst = outp + (long)n * DPAD * DPAD;
  for (int idx = t; idx < DPAD * DPAD; idx += 256) {
    int i = idx / DPAD, j = idx % DPAD;
    float v = 0.0f;
    if (i < 137 && j < 137) {
      v = Lsh[i * STR + j] / tr;
      if (i == j) v += tr * 1e-6f;
    }
    dst[idx] = v;
  }
}

// ---------------------------------------------------------------------------
// BiMap via fp32 WMMA: out[n] = Wb (64x137) @ outp (137x137) @ Wb^T.
// Padded K = 144; tmp (64x144) staged in LDS between the two GEMMs.
// ---------------------------------------------------------------------------
__global__ __launch_bounds__(256) void bimap_kernel(const float* __restrict__ outp,
                                                    const float* __restrict__ Wb,
                                                    float* __restrict__ out) {
  __shared__ float tmp[64][148];
  int t = threadIdx.x, w = t >> 5, lane = t & 31;
  int n = blockIdx.x;
  int m = lane & 15;
  int kb = (lane >> 4) * 2;
  int mb = (lane >> 4) * 8;
  const float* op = outp + (long)n * DPAD * DPAD;
  // stage 1: tmp = Wb_pad @ outp
  for (int tile = w; tile < 36; tile += 8) {
    int mi = tile / 9, nj = tile % 9;
    v8f acc = {0.f, 0.f, 0.f, 0.f, 0.f, 0.f, 0.f, 0.f};
    for (int k0 = 0; k0 < DPAD; k0 += 4) {
      int ka = k0 + kb;
      v2f a, b;
      a.x = (ka < 137) ? Wb[(mi * 16 + m) * 137 + ka] : 0.0f;
      a.y = (ka + 1 < 137) ? Wb[(mi * 16 + m) * 137 + ka + 1] : 0.0f;
      b.x = op[(long)ka * DPAD + nj * 16 + m];
      b.y = op[(long)(ka + 1) * DPAD + nj * 16 + m];
      acc = __builtin_amdgcn_wmma_f32_16x16x4_f32(false, a, false, b,
                                                  (short)0, acc, false, false);
    }
    for (int v = 0; v < 8; ++v) tmp[mi * 16 + mb + v][nj * 16 + m] = acc[v];
  }
  __syncthreads();
  // stage 2: out = tmp @ Wb_pad^T
  for (int tile = w; tile < 16; tile += 8) {
    int mi = tile / 4, nj = tile % 4;
    v8f acc = {0.f, 0.f, 0.f, 0.f, 0.f, 0.f, 0.f, 0.f};
    for (int k0 = 0; k0 < DPAD; k0 += 4) {
      int ka = k0 + kb;
      v2f a, b;
      a.x = tmp[mi * 16 + m][ka];
      a.y = tmp[mi * 16 + m][ka + 1];
      b.x = (ka < 137) ? Wb[(nj * 16 + m) * 137 + ka] : 0.0f;
      b.y = (ka + 1 < 137) ? Wb[(nj * 16 + m) * 137 + ka + 1] : 0.0f;
      acc = __builtin_amdgcn_wmma_f32_16x16x4_f32(false, a, false, b,
                                                  (short)0, acc, false, false);
    }
    for (int v = 0; v < 8; ++v)
      out[(long)n * 4096 + (mi * 16 + mb + v) * 64 + nj * 16 + m] = acc[v];
  }
}

// ---------------------------------------------------------------------------
extern "C" void kernel_launch(void* const* d_in, const int* in_sizes, int n_in,
                              void* d_out, int out_size, void* d_ws, size_t ws_size,
                              hipStream_t stream) {
  (void)in_sizes; (void)n_in; (void)out_size; (void)ws_size;
  const float* x  = (const float*)d_in[0];   // [32,2,64,25,16,16]
  const float* W  = (const float*)d_in[1];   // [16,16]
  const float* Wb = (const float*)d_in[2];   // [64,137]
  float* out = (float*)d_out;                // [32,64,64]

  char* ws = (char*)d_ws;
  size_t off = 0;
  auto wsalloc = [&](size_t bytes) -> float* {
    float* p = (float*)(ws + off);
    off = (off + bytes + 255) & ~(size_t)255;
    return p;
  };
  float* logx    = wsalloc((size_t)NN * PP * 256 * sizeof(float));     // 105 MB
  float* xc      = wsalloc((size_t)NN * PP * DPAD * sizeof(float));    // 59 MB
  float* meansum = wsalloc((size_t)NN * 256 * sizeof(float));
  float* sF      = wsalloc((size_t)(16 + 256) * sizeof(float));
  float* xm      = wsalloc((size_t)NN * DPAD * sizeof(float));
  float* xcov    = wsalloc((size_t)NN * DPAD * DPAD * sizeof(float));
  float* outp    = wsalloc((size_t)NN * DPAD * DPAD * sizeof(float));

  prep_kernel<<<1, 256, 0, stream>>>(W, sF, meansum);
  eig_logm_kernel<<<12800, 256, 0, stream>>>(x, logx, meansum);
  mean_logm_kernel<<<NN, 32, 0, stream>>>(meansum, xm);
  center_kernel<<<(NN * PP) / 2, 288, 0, stream>>>(logx, meansum, sF, xc);
  gram_kernel<<<dim3(9, NN), dim3(288), 0, stream>>>(xc, xcov);
  chol_kernel<<<NN, 256, (137 * 138 + 2) * sizeof(float), stream>>>(xcov, xm, outp);
  bimap_kernel<<<NN, 256, 0, stream>>>(outp, Wb, out);
}